// KBestDetector_55018531062640
// MI455X (gfx1250) — compile-verified
//
#include <hip/hip_runtime.h>

typedef __attribute__((ext_vector_type(2))) float v2f;
typedef __attribute__((ext_vector_type(8))) float v8f;

#define BIGF 1e30f
#define LLRCLIP 20.0f

// D += A(16x16) @ B(16x16), fp32, via 4 chained V_WMMA_F32_16X16X4_F32 (K-chunks of 4).
// A layout (16x4 f32, ISA 7.12.2): lane<16 -> row=lane, {x,y}={K0,K1}; lane>=16 -> row=lane-16, {K2,K3}
// B layout (4x16): VGPR0: lane<16 -> (K0,N=lane), lane>=16 -> (K2,N=lane-16); VGPR1: K1/K3
__device__ __forceinline__ v8f mm16_acc(const float* A, const float* B,
                                        v8f acc, bool negA, int lane) {
  const int hf = lane >> 4;
  const int r  = lane & 15;
#pragma unroll
  for (int q = 0; q < 4; ++q) {
    const int k0 = 4 * q + 2 * hf;
    v2f a, bb;
    a.x = A[r * 17 + k0];
    a.y = A[r * 17 + k0 + 1];
    if (negA) { a.x = -a.x; a.y = -a.y; }   // f32 WMMA has no A/B NEG modifier (ISA 7.12)
    bb.x = B[k0 * 17 + r];
    bb.y = B[(k0 + 1) * 17 + r];
    acc = __builtin_amdgcn_wmma_f32_16x16x4_f32(false, a, false, bb, (short)0, acc,
                                                false, false);
  }
  return acc;
}

__global__ __launch_bounds__(256) void kbest_kernel(
    const float* __restrict__ y_re, const float* __restrict__ y_im,
    const float* __restrict__ h_re, const float* __restrict__ h_im,
    const float* __restrict__ s_re, const float* __restrict__ s_im,
    const float* __restrict__ p_re, const float* __restrict__ p_im,
    float* __restrict__ out) {
  __shared__ float Lre[16][17], Lim[16][17];   // s -> Cholesky L (lower)
  __shared__ float Vre[16][17], Vim[16][17];   // L^-1
  __shared__ float Xre[16][17], Xim[16][17];   // [h | y | 0]
  __shared__ float Wre[16][17], Wim[16][17];   // [hw | yw] = L^-1 [h|y]
  __shared__ float Qre[16][9],  Qim[16][9];
  __shared__ float Rre[8][9],   Rim[8][9];
  __shared__ float ybr[8], ybi[8];
  __shared__ float nrm[8];
  __shared__ int   perm[8], invp[8];
  __shared__ float dist[64];
  __shared__ unsigned symp[64];                // 8 streams x 4-bit symbol, packed
  __shared__ float    key[1024];
  __shared__ unsigned val[1024];               // (parent<<8)|sym
  __shared__ float ptsR[16], ptsI[16];
  __shared__ float redR[16], redI[16];
  __shared__ float bcR, bcI;

  const int tid = threadIdx.x;
  const int b   = blockIdx.x;

  // ---------------- load (one block = one batch) ----------------
  {
    int i = tid >> 4, j = tid & 15;
    Lre[i][j] = s_re[b * 256 + tid];
    Lim[i][j] = s_im[b * 256 + tid];
    float xr = 0.f, xi = 0.f;
    if (j < 8) {
      xr = h_re[(b * 16 + i) * 8 + j];
      xi = h_im[(b * 16 + i) * 8 + j];
    } else if (j == 8) {
      xr = y_re[b * 16 + i];
      xi = y_im[b * 16 + i];
    }
    Xre[i][j] = xr; Xim[i][j] = xi;
    if (tid < 16) { ptsR[tid] = p_re[tid]; ptsI[tid] = p_im[tid]; }
  }
  __syncthreads();

  // ---------------- complex Cholesky of s (in place, lower) ----------------
  for (int j = 0; j < 16; ++j) {
    if (tid == 0) {
      float d = Lre[j][j];
      for (int p = 0; p < j; ++p) d -= Lre[j][p] * Lre[j][p] + Lim[j][p] * Lim[j][p];
      Lre[j][j] = sqrtf(fmaxf(d, 1e-20f));
      Lim[j][j] = 0.f;
    }
    __syncthreads();
    if (tid > j && tid < 16) {
      int i = tid;
      float ar = Lre[i][j], ai = Lim[i][j];
      for (int p = 0; p < j; ++p) {
        float br = Lre[i][p], bi = Lim[i][p];
        float cr = Lre[j][p], ci = Lim[j][p];
        ar -= br * cr + bi * ci;               // -= L[i][p] * conj(L[j][p])
        ai -= bi * cr - br * ci;
      }
      float inv = 1.f / Lre[j][j];
      Lre[i][j] = ar * inv;
      Lim[i][j] = ai * inv;
    }
    __syncthreads();
  }

  // ---------------- triangular inverse: V = L^-1 (thread c -> column c) ----
  if (tid < 16) {
    int c = tid;
    for (int i = 0; i < c; ++i) { Vre[i][c] = 0.f; Vim[i][c] = 0.f; }
    Vre[c][c] = 1.f / Lre[c][c];
    Vim[c][c] = 0.f;
    for (int i = c + 1; i < 16; ++i) {
      float sr = 0.f, si = 0.f;
      for (int p = c; p < i; ++p) {
        sr += Lre[i][p] * Vre[p][c] - Lim[i][p] * Vim[p][c];
        si += Lre[i][p] * Vim[p][c] + Lim[i][p] * Vre[p][c];
      }
      float inv = 1.f / Lre[i][i];
      Vre[i][c] = -sr * inv;
      Vim[i][c] = -si * inv;
    }
  }
  __syncthreads();

  // ---------------- whitening GEMM on the matrix core (wave 0, EXEC all-1s) --
  // [hw|yw] = V * X   (complex): Re = Vr*Xr - Vi*Xi ; Im = Vr*Xi + Vi*Xr
  if (tid < 32) {
    v8f cR = {0, 0, 0, 0, 0, 0, 0, 0};
    v8f cI = {0, 0, 0, 0, 0, 0, 0, 0};
    cR = mm16_acc(&Vre[0][0], &Xre[0][0], cR, false, tid);
    cR = mm16_acc(&Vim[0][0], &Xim[0][0], cR, true,  tid);
    cI = mm16_acc(&Vre[0][0], &Xim[0][0], cI, false, tid);
    cI = mm16_acc(&Vim[0][0], &Xre[0][0], cI, false, tid);
    int hf = tid >> 4, r = tid & 15;
#pragma unroll
    for (int v = 0; v < 8; ++v) {              // C/D layout: VGPR v -> M = v + 8*hf, N = r
      Wre[v + 8 * hf][r] = cR[v];
      Wim[v + 8 * hf][r] = cI[v];
    }
  }
  __syncthreads();

  // ---------------- column norms + descending argsort ----------------
  if (tid < 8) {
    float a = 0.f;
    for (int m = 0; m < 16; ++m) a += Wre[m][tid] * Wre[m][tid] + Wim[m][tid] * Wim[m][tid];
    nrm[tid] = a;
  }
  __syncthreads();
  if (tid == 0) {
    int idx[8];
    for (int i = 0; i < 8; ++i) idx[i] = i;
    for (int i = 0; i < 8; ++i) {
      int best = i;
      for (int j = i + 1; j < 8; ++j)
        if (nrm[idx[j]] > nrm[idx[best]]) best = j;
      int t = idx[i]; idx[i] = idx[best]; idx[best] = t;
      perm[i] = idx[i];
    }
    for (int i = 0; i < 8; ++i) invp[perm[i]] = i;
  }
  __syncthreads();

  // ---------------- MGS QR of permuted hw (rows on tid<16) ----------------
  {
    int m = tid & 15;
    float vr = 0.f, vi = 0.f;
    for (int j = 0; j < 8; ++j) {
      int cj = perm[j];
      if (tid < 16) { vr = Wre[m][cj]; vi = Wim[m][cj]; }
      for (int i = 0; i < j; ++i) {
        if (tid < 16) {
          float qr = Qre[m][i], qi = Qim[m][i];
          redR[m] = qr * vr + qi * vi;         // conj(q) * v
          redI[m] = qr * vi - qi * vr;
        }
        __syncthreads();
        if (tid == 0) {
          float pr = 0.f, pi = 0.f;
          for (int t = 0; t < 16; ++t) { pr += redR[t]; pi += redI[t]; }
          Rre[i][j] = pr; Rim[i][j] = pi; bcR = pr; bcI = pi;
        }
        __syncthreads();
        if (tid < 16) {
          float pr = bcR, pi = bcI;
          float qr = Qre[m][i], qi = Qim[m][i];
          vr -= pr * qr - pi * qi;
          vi -= pr * qi + pi * qr;
        }
        __syncthreads();
      }
      if (tid < 16) redR[m] = vr * vr + vi * vi;
      __syncthreads();
      if (tid == 0) {
        float nn = 0.f;
        for (int t = 0; t < 16; ++t) nn += redR[t];
        float rjj = sqrtf(fmaxf(nn, 0.f));
        Rre[j][j] = rjj; Rim[j][j] = 0.f;
        bcR = (rjj > 0.f) ? 1.f / rjj : 0.f;
      }
      __syncthreads();
      if (tid < 16) { Qre[m][j] = vr * bcR; Qim[m][j] = vi * bcR; }
      __syncthreads();
    }
    // ybar[i] = q_i^H yw  (yw = column 8 of W)
    for (int i = 0; i < 8; ++i) {
      if (tid < 16) {
        float qr = Qre[m][i], qi = Qim[m][i];
        float yr = Wre[m][8], yi = Wim[m][8];
        redR[m] = qr * yr + qi * yi;
        redI[m] = qr * yi - qi * yr;
      }
      __syncthreads();
      if (tid == 0) {
        float pr = 0.f, pi = 0.f;
        for (int t = 0; t < 16; ++t) { pr += redR[t]; pi += redI[t]; }
        ybr[i] = pr; ybi[i] = pi;
      }
      __syncthreads();
    }
  }

  // ---------------- K-best tree search (levels 7..0) ----------------
  if (tid < 64) {
    dist[tid] = (tid == 0) ? 0.f : BIGF;
    symp[tid] = 0u;
  }
  __syncthreads();

  for (int lev = 7; lev >= 0; --lev) {
    // expand: 64 parents x 16 symbols = 1024 candidates (4 per thread, same parent)
    {
      int p = tid >> 2;
      float dp = dist[p];
      unsigned ps = symp[p];
      float zr = ybr[lev], zi = ybi[lev];
      for (int j = lev + 1; j < 8; ++j) {
        int cs = (int)((ps >> (4 * j)) & 15u);
        float rr = Rre[lev][j], ri = Rim[lev][j];
        float xr = ptsR[cs], xi = ptsI[cs];
        zr -= rr * xr - ri * xi;
        zi -= rr * xi + ri * xr;
      }
      float rii = Rre[lev][lev];               // real by construction
#pragma unroll
      for (int q = 0; q < 4; ++q) {
        int c = ((tid & 3) << 2) + q;
        float dr = zr - rii * ptsR[c];
        float di = zi - rii * ptsI[c];
        key[p * 16 + c] = dp + dr * dr + di * di;
        val[p * 16 + c] = ((unsigned)p << 8) | (unsigned)c;
      }
    }
    __syncthreads();

    // bitonic sort 1024 ascending (keys + payload), 256 threads
    for (int size = 2; size <= 1024; size <<= 1) {
      for (int strd = size >> 1; strd > 0; strd >>= 1) {
        for (int i = tid; i < 1024; i += 256) {
          int j = i ^ strd;
          if (j > i) {
            bool asc = ((i & size) == 0);
            float ki = key[i], kj = key[j];
            if (asc ? (ki > kj) : (ki < kj)) {
              key[i] = kj; key[j] = ki;
              unsigned tv = val[i]; val[i] = val[j]; val[j] = tv;
            }
          }
        }
        __syncthreads();
      }
    }

    // keep the 64 best, extend symbol paths
    float nd = 0.f; unsigned nsy = 0u;
    if (tid < 64) {
      nd = key[tid];
      unsigned v = val[tid];
      unsigned p = v >> 8, c = v & 15u;
      unsigned old = symp[p];
      nsy = (old & ~(0xFu << (4 * lev))) | (c << (4 * lev));
    }
    __syncthreads();
    if (tid < 64) { dist[tid] = nd; symp[tid] = nsy; }
    __syncthreads();
  }

  // ---------------- LLRs: 8 streams x 4 bits (MSB-first), clip +-20 ---------
  if (tid < 32) {
    int s = tid >> 2, bit = tid & 3;
    int j = invp[s];                            // sorted position of original stream s
    float d0 = BIGF, d1 = BIGF;
    for (int kk = 0; kk < 64; ++kk) {
      int c = (int)((symp[kk] >> (4 * j)) & 15u);
      float d = dist[kk];
      if ((c >> (3 - bit)) & 1) d1 = fminf(d1, d);
      else                      d0 = fminf(d0, d);
    }
    float llr = fminf(fmaxf(d0 - d1, -LLRCLIP), LLRCLIP);
    out[((b * 8) + s) * 4 + bit] = llr;
  }
}

extern "C" void kernel_launch(void* const* d_in, const int* in_sizes, int n_in,
                              void* d_out, int out_size, void* d_ws, size_t ws_size,
                              hipStream_t stream) {
  (void)n_in; (void)out_size; (void)d_ws; (void)ws_size;
  const float* y_re = (const float*)d_in[0];
  const float* y_im = (const float*)d_in[1];
  const float* h_re = (const float*)d_in[2];
  const float* h_im = (const float*)d_in[3];
  const float* s_re = (const float*)d_in[4];
  const float* s_im = (const float*)d_in[5];
  const float* p_re = (const float*)d_in[6];
  const float* p_im = (const float*)d_in[7];
  // d_in[8] = k (fixed 64; encoded in the kernel's data structures)
  int B = in_sizes[0] / 16;                     // y_re is (B, 16)
  kbest_kernel<<<B, 256, 0, stream>>>(y_re, y_im, h_re, h_im, s_re, s_im,
                                      p_re, p_im, (float*)d_out);
}